// DirectSolverNet_42863773614738
// MI455X (gfx1250) — compile-verified
//
#include <hip/hip_runtime.h>
#include <math.h>

// Problem dims
#define B_   32
#define C_   8
#define H_   120
#define Wd_  160
#define HW_  (H_ * Wd_)         // 19200
#define CHW_ (C_ * HW_)         // 153600
#define NS_  10

typedef float v2f __attribute__((ext_vector_type(2)));
typedef float v8f __attribute__((ext_vector_type(8)));

// Workspace layout (float offsets)
#define WS_JTR   0              // B*6        = 192
#define WS_JTRS  192            // B*6*NS     = 1920
#define WS_RS    2112           // B*NS*9     = 2880
#define WS_TS    4992           // B*NS*3     = 960
#define WS_DAMP  5952           // B*6        = 192
#define WS_ZERO_N 2112          // jtr + jtrs must be zeroed

__device__ __forceinline__ v8f wmma4(v2f a, v2f b, v8f c) {
  // V_WMMA_F32_16X16X4_F32 : D = A(16x4) * B(4x16) + C(16x16), all f32
  return __builtin_amdgcn_wmma_f32_16x16x4_f32(false, a, false, b, (short)0, c,
                                               false, false);
}

// ---------------- small per-thread linear algebra ----------------
__device__ void solve6(float A[6][6], float b[6], float x[6]) {
  for (int k = 0; k < 6; ++k) {
    int piv = k; float mx = fabsf(A[k][k]);
    for (int r = k + 1; r < 6; ++r) { float v = fabsf(A[r][k]); if (v > mx) { mx = v; piv = r; } }
    if (piv != k) {
      for (int c = k; c < 6; ++c) { float t = A[k][c]; A[k][c] = A[piv][c]; A[piv][c] = t; }
      float t = b[k]; b[k] = b[piv]; b[piv] = t;
    }
    float inv = 1.0f / A[k][k];
    for (int r = k + 1; r < 6; ++r) {
      float f = A[r][k] * inv;
      for (int c = k; c < 6; ++c) A[r][c] -= f * A[k][c];
      b[r] -= f * b[k];
    }
  }
  for (int r = 5; r >= 0; --r) {
    float s = b[r];
    for (int c = r + 1; c < 6; ++c) s -= A[r][c] * x[c];
    x[r] = s / A[r][r];
  }
}

__device__ void pose_from_xi(const float* __restrict__ R0, const float* __restrict__ t0,
                             const float xi[6], float* __restrict__ Rout, float* __restrict__ tout) {
  float tw0 = -xi[3], tw1 = -xi[4], tw2 = -xi[5];
  float theta = sqrtf(tw0 * tw0 + tw1 * tw1 + tw2 * tw2);
  float w0 = tw0 / theta, w1 = tw1 / theta, w2 = tw2 / theta;
  float s = sinf(theta), cc = 1.0f - cosf(theta);
  float dR[9];
  dR[0] = 1.0f + cc * (-(w1 * w1 + w2 * w2));
  dR[1] = s * (-w2) + cc * (w0 * w1);
  dR[2] = s * ( w1) + cc * (w0 * w2);
  dR[3] = s * ( w2) + cc * (w0 * w1);
  dR[4] = 1.0f + cc * (-(w0 * w0 + w2 * w2));
  dR[5] = s * (-w0) + cc * (w1 * w2);
  dR[6] = s * (-w1) + cc * (w0 * w2);
  dR[7] = s * ( w0) + cc * (w1 * w2);
  dR[8] = 1.0f + cc * (-(w0 * w0 + w1 * w1));
  float dt0 = -(dR[0] * xi[0] + dR[1] * xi[1] + dR[2] * xi[2]);
  float dt1 = -(dR[3] * xi[0] + dR[4] * xi[1] + dR[5] * xi[2]);
  float dt2 = -(dR[6] * xi[0] + dR[7] * xi[1] + dR[8] * xi[2]);
  for (int r = 0; r < 3; ++r) {
    for (int c = 0; c < 3; ++c)
      Rout[r * 3 + c] = R0[r * 3 + 0] * dR[0 + c] + R0[r * 3 + 1] * dR[3 + c] + R0[r * 3 + 2] * dR[6 + c];
    tout[r] = R0[r * 3 + 0] * dt0 + R0[r * 3 + 1] * dt1 + R0[r * 3 + 2] * dt2 + t0[r];
  }
}

// ---------------- K0: zero accumulators ----------------
__global__ void k0_zero(float* __restrict__ p, int n) {
  int i = blockIdx.x * blockDim.x + threadIdx.x;
  if (i < n) p[i] = 0.0f;
}

// ---------------- K1: JtR = Jt @ (weights*resid) ----------------
__global__ void __launch_bounds__(256) k1_jtr(const float* __restrict__ Jt,
                                              const float* __restrict__ weights,
                                              const float* __restrict__ resid,
                                              float* __restrict__ jtr) {
  __shared__ float red[8 * 6];
  int b = blockIdx.x;
  int tid = threadIdx.x;
  const float* Jb = Jt + (size_t)b * 6 * CHW_;
  const float* wb = weights + (size_t)b * CHW_;
  const float* rb = resid + (size_t)b * CHW_;
  float acc[6] = {0, 0, 0, 0, 0, 0};
  int stride = gridDim.y * blockDim.x;
  for (int n = blockIdx.y * blockDim.x + tid; n < CHW_; n += stride) {
    float wr = wb[n] * rb[n];
#pragma unroll
    for (int i = 0; i < 6; ++i) acc[i] += Jb[(size_t)i * CHW_ + n] * wr;
  }
#pragma unroll
  for (int i = 0; i < 6; ++i)
    for (int off = 16; off; off >>= 1) acc[i] += __shfl_down(acc[i], off, 32);
  int wave = tid >> 5, lane = tid & 31;
  if (lane == 0)
    for (int i = 0; i < 6; ++i) red[wave * 6 + i] = acc[i];
  __syncthreads();
  if (tid < 6) {
    float s = 0.0f;
    for (int w = 0; w < 8; ++w) s += red[w * 6 + tid];
    atomicAdd(&jtr[b * 6 + tid], s);
  }
}

// ---------------- K2: per (b, lambda) damped solve -> pose candidates ----------------
__global__ void k2_poses(const float* __restrict__ JtJ, const float* __restrict__ R0,
                         const float* __restrict__ t0, const float* __restrict__ jtr,
                         float* __restrict__ Rs, float* __restrict__ ts) {
  int t = blockIdx.x * blockDim.x + threadIdx.x;
  if (t >= B_ * NS_) return;
  int b = t / NS_, s = t % NS_;
  float lam = powf(10.0f, -5.0f + (10.0f / 9.0f) * (float)s);
  const float* J = JtJ + b * 36;
  float trace = J[0] + J[7] + J[14] + J[21] + J[28] + J[35];
  float eps = trace * 1e-6f;
  float A[6][6], rhs[6], xi[6];
  for (int i = 0; i < 6; ++i)
    for (int j = 0; j < 6; ++j)
      A[i][j] = J[i * 6 + j] + ((i == j) ? (lam * J[i * 6 + j] + eps) : 0.0f);
  for (int i = 0; i < 6; ++i) rhs[i] = jtr[b * 6 + i];
  solve6(A, rhs, xi);
  pose_from_xi(R0 + b * 9, t0 + b * 3, xi, Rs + ((size_t)b * NS_ + s) * 9, ts + ((size_t)b * NS_ + s) * 3);
}

// ---------------- K3: fused warped residual (all 10 lambdas) + WMMA Jt reduction ----------------
#define K3_WAVES 4
#define K3_BLKY  30
#define CHUNKS_  (HW_ / 32)     // 600
#define WRS 11                  // odd LDS stride -> conflict free

__global__ void __launch_bounds__(128) k3_main(
    const float* __restrict__ Jt, const float* __restrict__ weights,
    const float* __restrict__ invD0, const float* __restrict__ invD1,
    const float* __restrict__ x0, const float* __restrict__ x1,
    const float* __restrict__ Kc, const float* __restrict__ Rs,
    const float* __restrict__ ts, float* __restrict__ jtrs) {
  __shared__ float lds_w[K3_WAVES][8 * 32 * WRS + 16];
  __shared__ float lds_red[K3_WAVES][64];

  int b = blockIdx.x;
  int tid = threadIdx.x;
  int wave = __builtin_amdgcn_readfirstlane(tid >> 5);  // wave-uniform -> scalar control flow
  int lane = tid & 31;
  int row = lane & 15;                 // A: M row (Jt row) / B: N col (lambda)
  int hi = lane >> 4;
  int rowSafe = row < 6 ? row : 5;     // clamp: rows 6..15 replicate row 5 (only M<6 is read out)

  float fx = Kc[b * 4 + 0], fy = Kc[b * 4 + 1], cx = Kc[b * 4 + 2], cy = Kc[b * 4 + 3];
  const float* x1b = x1 + (size_t)b * CHW_;
  const float* d1b = invD1 + (size_t)b * HW_;
  const float* Jb = Jt + (size_t)b * 6 * CHW_;
  float* myw = lds_w[wave];

  // zero the per-row pad slot (read by the unused N=10 B-column) once
  for (int i = lane; i < 8 * 32; i += 32) myw[i * WRS + NS_] = 0.0f;

  v8f acc = {};
  const int wstride = K3_BLKY * K3_WAVES;  // 120 waves per batch
  for (int ch = blockIdx.y * K3_WAVES + wave; ch < CHUNKS_; ch += wstride) {
    // prefetch next chunk's Jt (c=0 plane) -> global_prefetch_b8
    int nch = ch + wstride;
    if (nch < CHUNKS_)
      __builtin_prefetch(Jb + (size_t)rowSafe * CHW_ + nch * 32, 0, 1);

    int pix = ch * 32 + lane;
    int ph = pix / Wd_;
    int pw = pix - ph * Wd_;
    float px = ((float)pw - cx) / fx;
    float py = ((float)ph - cy) / fy;
    float d0 = invD0[(size_t)b * HW_ + pix];
    float x0v[8], wv[8];
#pragma unroll
    for (int c = 0; c < 8; ++c) {
      x0v[c] = x0[(size_t)b * CHW_ + c * HW_ + pix];
      wv[c] = weights[(size_t)b * CHW_ + c * HW_ + pix];
    }
    // Phase 1: weighted residuals for all lambdas, staged in LDS
    for (int s = 0; s < NS_; ++s) {
      const float* Rp = Rs + ((size_t)b * NS_ + s) * 9;
      const float* tp = ts + ((size_t)b * NS_ + s) * 3;
      float X = Rp[0] * px + Rp[1] * py + Rp[2] + tp[0] * d0;
      float Y = Rp[3] * px + Rp[4] * py + Rp[5] + tp[1] * d0;
      float Z = Rp[6] * px + Rp[7] * py + Rp[8] + tp[2] * d0;
      float u_ = X / Z * fx + cx;
      float v_ = Y / Z * fy + cy;
      float inv_z = d0 / Z;
      float uc = fminf(fmaxf(u_, 0.0f), (float)(Wd_ - 1));
      float vc = fminf(fmaxf(v_, 0.0f), (float)(H_ - 1));
      float u0f = floorf(uc), v0f = floorf(vc);
      float au = uc - u0f, av = vc - v0f;
      int iu0 = (int)u0f, iv0 = (int)v0f;
      int iu1 = iu0 < (Wd_ - 1) ? iu0 + 1 : (Wd_ - 1);
      int iv1 = iv0 < (H_ - 1) ? iv0 + 1 : (H_ - 1);
      int i00 = iv0 * Wd_ + iu0, i01 = iv0 * Wd_ + iu1;
      int i10 = iv1 * Wd_ + iu0, i11 = iv1 * Wd_ + iu1;
      float w00 = (1.0f - av) * (1.0f - au), w01 = (1.0f - av) * au;
      float w10 = av * (1.0f - au), w11 = av * au;
      float dw = w00 * d1b[i00] + w01 * d1b[i01] + w10 * d1b[i10] + w11 * d1b[i11];
      bool inview = (inv_z > dw - 0.1f) && (u_ > 0.0f) && (u_ < (float)Wd_) &&
                    (v_ > 0.0f) && (v_ < (float)H_);
#pragma unroll
      for (int c = 0; c < 8; ++c) {
        const float* xc = x1b + c * HW_;
        float x1w = w00 * xc[i00] + w01 * xc[i01] + w10 * xc[i10] + w11 * xc[i11];
        float res = inview ? (x1w - x0v[c]) : 0.001f;
        myw[(c * 32 + lane) * WRS + s] = wv[c] * res;
      }
    }
    // Phase 2: WMMA f32 16x16x4 accumulation over this chunk (all 8 channels)
#pragma unroll
    for (int c = 0; c < 8; ++c) {
      const float* Arow = Jb + (size_t)rowSafe * CHW_ + c * HW_ + ch * 32;
#pragma unroll
      for (int j = 0; j < 8; ++j) {
        int koff = 4 * j + 2 * hi;  // 16x4 f32 A layout: lanes 0-15 hold K=0,1 ; lanes 16-31 K=2,3
        v2f Af = *(const v2f*)(Arow + koff);
        v2f Bf;
        Bf.x = myw[(c * 32 + koff + 0) * WRS + row];
        Bf.y = myw[(c * 32 + koff + 1) * WRS + row];
        acc = wmma4(Af, Bf, acc);
      }
    }
  }
  // C/D layout: lanes 0-15 -> N=lane, VGPR r -> M=r. Useful: M<6 (r<6), N<10 (lane<10).
  if (lane < 10) {
#pragma unroll
    for (int r = 0; r < 6; ++r) lds_red[wave][r * 10 + lane] = acc[r];
  }
  __syncthreads();
  if (tid < 60) {
    float s = 0.0f;
#pragma unroll
    for (int w = 0; w < K3_WAVES; ++w) s += lds_red[w][tid];
    atomicAdd(&jtrs[b * 60 + tid], s);  // jtrs[b, i*10 + s]
  }
}

// ---------------- K4: MLP (feat 32x96 -> 128 -> 256 -> 6) via WMMA ----------------
__device__ __forceinline__ void mlp_layer(const float* __restrict__ Alds, int lda,
                                          const float* __restrict__ Wg, int Ndim, int Kdim,
                                          const float* __restrict__ bias,
                                          float* __restrict__ Cmem, int ldc, int tid) {
  int wave = __builtin_amdgcn_readfirstlane(tid >> 5);
  int lane = tid & 31;
  int row = lane & 15, hi = lane >> 4;
  int Ntiles = (Ndim + 15) >> 4;
  int T = 2 * Ntiles;
  for (int t = wave; t < T; t += 8) {
    int mt = t & 1, nt = t >> 1;
    int ncol = nt * 16 + row;
    bool nv = ncol < Ndim;
    int ncolSafe = nv ? ncol : 0;  // clamp: garbage only lands in unused N columns
    v8f acc = {};
#pragma unroll 4
    for (int k0 = 0; k0 < Kdim; k0 += 4) {
      int kk = k0 + 2 * hi;
      const float* ar = Alds + (mt * 16 + row) * lda + kk;
      v2f Af; Af.x = ar[0]; Af.y = ar[1];
      v2f Bf;
      Bf.x = Wg[kk * Ndim + ncolSafe];
      Bf.y = Wg[(kk + 1) * Ndim + ncolSafe];
      acc = wmma4(Af, Bf, acc);
    }
    if (nv) {
      float bv = bias[ncol];
#pragma unroll
      for (int r = 0; r < 8; ++r) {
        int m = mt * 16 + hi * 8 + r;
        Cmem[m * ldc + ncol] = fmaxf(acc[r] + bv, 0.0f);
      }
    }
  }
}

__global__ void __launch_bounds__(256) k4_mlp(const float* __restrict__ JtJ,
                                              const float* __restrict__ jtrs,
                                              const float* __restrict__ W1, const float* __restrict__ b1,
                                              const float* __restrict__ W2, const float* __restrict__ b2,
                                              const float* __restrict__ W3, const float* __restrict__ b3,
                                              float* __restrict__ damp) {
  __shared__ float feat[32 * 97];
  __shared__ float h1[32 * 129];
  __shared__ float h2[32 * 257];
  int tid = threadIdx.x;
  for (int idx = tid; idx < 32 * 96; idx += 256) {
    int bb = idx / 96, f = idx - bb * 96;
    feat[bb * 97 + f] = (f < 60) ? jtrs[bb * 60 + f] : JtJ[bb * 36 + (f - 60)];
  }
  __syncthreads();
  mlp_layer(feat, 97, W1, 128, 96, b1, h1, 129, tid);
  __syncthreads();
  mlp_layer(h1, 129, W2, 256, 128, b2, h2, 257, tid);
  __syncthreads();
  mlp_layer(h2, 257, W3, 6, 256, b3, damp, 6, tid);
}

// ---------------- K5: final damped solve -> output pose ----------------
__global__ void k5_final(const float* __restrict__ JtJ, const float* __restrict__ R0,
                         const float* __restrict__ t0, const float* __restrict__ jtr,
                         const float* __restrict__ damp, float* __restrict__ out) {
  int b = threadIdx.x;
  if (b >= B_) return;
  const float* J = JtJ + b * 36;
  float trace = J[0] + J[7] + J[14] + J[21] + J[28] + J[35];
  float eps = trace * 1e-6f;
  float A[6][6], rhs[6], xi[6];
  for (int i = 0; i < 6; ++i)
    for (int j = 0; j < 6; ++j)
      A[i][j] = J[i * 6 + j] + ((i == j) ? (damp[b * 6 + i] + eps) : 0.0f);
  for (int i = 0; i < 6; ++i) rhs[i] = jtr[b * 6 + i];
  solve6(A, rhs, xi);
  float Rn[9], tn[3];
  pose_from_xi(R0 + b * 9, t0 + b * 3, xi, Rn, tn);
  for (int k = 0; k < 9; ++k) out[b * 9 + k] = Rn[k];
  for (int k = 0; k < 3; ++k) out[B_ * 9 + b * 3 + k] = tn[k];
}

// ---------------- host launcher ----------------
extern "C" void kernel_launch(void* const* d_in, const int* in_sizes, int n_in,
                              void* d_out, int out_size, void* d_ws, size_t ws_size,
                              hipStream_t stream) {
  (void)in_sizes; (void)n_in; (void)out_size; (void)ws_size;
  const float* JtJ = (const float*)d_in[0];
  const float* Jt = (const float*)d_in[1];
  const float* weights = (const float*)d_in[2];
  const float* resid = (const float*)d_in[3];
  const float* pose0_R = (const float*)d_in[4];
  const float* pose0_t = (const float*)d_in[5];
  const float* invD0 = (const float*)d_in[6];
  const float* invD1 = (const float*)d_in[7];
  const float* x0 = (const float*)d_in[8];
  const float* x1 = (const float*)d_in[9];
  const float* Kc = (const float*)d_in[10];
  const float* W1 = (const float*)d_in[11];
  const float* b1 = (const float*)d_in[12];
  const float* W2 = (const float*)d_in[13];
  const float* b2 = (const float*)d_in[14];
  const float* W3 = (const float*)d_in[15];
  const float* b3 = (const float*)d_in[16];
  float* out = (float*)d_out;
  float* ws = (float*)d_ws;

  float* jtr  = ws + WS_JTR;
  float* jtrs = ws + WS_JTRS;
  float* Rs   = ws + WS_RS;
  float* ts   = ws + WS_TS;
  float* damp = ws + WS_DAMP;

  k0_zero<<<(WS_ZERO_N + 255) / 256, 256, 0, stream>>>(ws, WS_ZERO_N);
  k1_jtr<<<dim3(B_, 75), 256, 0, stream>>>(Jt, weights, resid, jtr);
  k2_poses<<<(B_ * NS_ + 63) / 64, 64, 0, stream>>>(JtJ, pose0_R, pose0_t, jtr, Rs, ts);
  k3_main<<<dim3(B_, K3_BLKY), 128, 0, stream>>>(Jt, weights, invD0, invD1, x0, x1, Kc, Rs, ts, jtrs);
  k4_mlp<<<1, 256, 0, stream>>>(JtJ, jtrs, W1, b1, W2, b2, W3, b3, damp);
  k5_final<<<1, 32, 0, stream>>>(JtJ, pose0_R, pose0_t, jtr, damp, out);
}